// ContrastLoss_54417235640831
// MI455X (gfx1250) — compile-verified
//
#include <hip/hip_runtime.h>
#include <math.h>

// ContrastLoss for MI455X (gfx1250, wave32, WMMA).
// feat: [4,64,512,512] f32, gt: [4,19,512,512] i32 -> scalar f32 loss.
// Bandwidth-bound; fp32 WMMA (V_WMMA_F32_16X16X4_F32) for both GEMMs.
// Pass 1 additionally bit-packs the gt mask (19 bits/pixel, 4MB) so pass 3
// avoids re-reading the 80MB gt tensor: ~700MB -> ~628MB total HBM traffic.

#define TAUV 0.07f
#define EPSV 1e-12f

typedef float v2f __attribute__((ext_vector_type(2)));
typedef float v8f __attribute__((ext_vector_type(8)));

static constexpr int Bb   = 4;
static constexpr int Cc   = 64;
static constexpr int Kk   = 19;
static constexpr int HW   = 512 * 512;
static constexpr int NPIX = Bb * HW;           // 1,048,576 pixels

__device__ __forceinline__ v8f wmma_f32_k4(v2f a, v2f b, v8f c) {
  // D[16x16] = A[16x4] * B[4x16] + C, all fp32.
  return __builtin_amdgcn_wmma_f32_16x16x4_f32(
      /*neg_a=*/false, a, /*neg_b=*/false, b,
      /*c_mod=*/(short)0, c, /*reuse_a=*/false, /*reuse_b=*/false);
}

// ---------------------------------------------------------------------------
// Kernel 0: zero workspace accumulators.
// ---------------------------------------------------------------------------
__global__ void zero_kernel(float* __restrict__ ws, int n) {
  int i = blockIdx.x * blockDim.x + threadIdx.x;
  if (i < n) ws[i] = 0.0f;
}

// ---------------------------------------------------------------------------
// Kernel 1: prototypes k0[k][c] = sum over pixels of mask[k,p]*feat[c,p],
// plus num_pos, plus (optionally) the bit-packed per-pixel class mask.
//   A tile: mask, M=k (19 padded to 32 -> 2 tiles), K = 4 pixels
//   B tile: feat, K = 4 pixels, N = channel (64 -> 4 tiles)
// Wave-private WMMA accumulators; reduced via LDS ds_add_f32 + global atomics.
// ---------------------------------------------------------------------------
template <bool WRITE_MASK>
__global__ void __launch_bounds__(256)
proto_kernel(const float* __restrict__ feat, const int* __restrict__ gt,
             float* __restrict__ k0_acc, unsigned int* __restrict__ npos,
             unsigned int* __restrict__ wsmask) {
  __shared__ float lds_k0[32 * 64];
  __shared__ unsigned int lds_cnt;
  const int tid = threadIdx.x;
  for (int i = tid; i < 32 * 64; i += 256) lds_k0[i] = 0.0f;
  if (tid == 0) lds_cnt = 0u;
  __syncthreads();

  const int lane = tid & 31;
  const int half = lane >> 4;        // which half-wave
  const int l16  = lane & 15;
  const int wave = blockIdx.x * 8 + (tid >> 5);
  const int nw   = gridDim.x * 8;

  v8f acc[2][4];
#pragma unroll
  for (int mt = 0; mt < 2; ++mt)
#pragma unroll
    for (int nt = 0; nt < 4; ++nt) acc[mt][nt] = (v8f)0.0f;
  unsigned int cnt = 0;

  const int nkb = NPIX / 4;          // 4-pixel K blocks
  for (int kb = wave; kb < nkb; kb += nw) {
    const int pg  = kb * 4;
    const int b   = pg / HW;
    const int pin = (pg - b * HW) + half * 2;   // lane's first pixel in batch

    // A (mask): 16x4 f32; lanes 0-15 hold K=0,1 in v0,v1; lanes 16-31 K=2,3.
    bool pe[2] = {false, false};   // predicate, even pixel of this half
    bool po[2] = {false, false};   // predicate, odd pixel of this half
#pragma unroll
    for (int mt = 0; mt < 2; ++mt) {
      const int k = mt * 16 + l16;
      if (k < Kk) {
        const int* gp = gt + (size_t)(b * Kk + k) * HW + pin;
        pe[mt] = (gp[0] == 1);
        po[mt] = (gp[1] == 1);
        cnt += (unsigned)pe[mt] + (unsigned)po[mt];
      }
    }
    v2f a[2];
#pragma unroll
    for (int mt = 0; mt < 2; ++mt) {
      v2f av; av.x = pe[mt] ? 1.0f : 0.0f; av.y = po[mt] ? 1.0f : 0.0f;
      a[mt] = av;
    }

    if (WRITE_MASK) {
      // Convergent ballots: bit i = predicate of lane i.
      // lanes 0-15 carry pixels {0,1}, lanes 16-31 carry pixels {2,3};
      // tile0 lanes map to k=0..15, tile1 lanes 0-2/16-18 map to k=16..18.
      const unsigned e0 = (unsigned)__ballot(pe[0]);
      const unsigned o0 = (unsigned)__ballot(po[0]);
      const unsigned e1 = (unsigned)__ballot(pe[1]);
      const unsigned o1 = (unsigned)__ballot(po[1]);
      const unsigned lo = (lane & 1) ? o0 : e0;
      const unsigned hi = (lane & 1) ? o1 : e1;
      const unsigned mw = (lane & 2)
          ? ((lo >> 16) | (((hi >> 16) & 7u) << 16))
          : ((lo & 0xffffu) | ((hi & 7u) << 16));
      if (lane < 4) wsmask[pg + lane] = mw;
    }

    // B (feat): 4x16 f32; lane = channel col, K rows split by half-wave.
    v2f bt[4];
#pragma unroll
    for (int nt = 0; nt < 4; ++nt) {
      const int c = nt * 16 + l16;
      const float* fp = feat + (size_t)(b * Cc + c) * HW + pin;
      v2f bv; bv.x = fp[0]; bv.y = fp[1];
      bt[nt] = bv;
    }
#pragma unroll
    for (int mt = 0; mt < 2; ++mt)
#pragma unroll
      for (int nt = 0; nt < 4; ++nt)
        acc[mt][nt] = wmma_f32_k4(a[mt], bt[nt], acc[mt][nt]);
  }

  // C/D layout: vgpr j, lanes 0-15 -> row j, lanes 16-31 -> row 8+j; col=l16.
#pragma unroll
  for (int mt = 0; mt < 2; ++mt)
#pragma unroll
    for (int nt = 0; nt < 4; ++nt)
#pragma unroll
      for (int j = 0; j < 8; ++j) {
        const int row = mt * 16 + half * 8 + j;
        const int col = nt * 16 + l16;
        atomicAdd(&lds_k0[row * 64 + col], acc[mt][nt][j]);
      }
  atomicAdd(&lds_cnt, cnt);
  __syncthreads();

  for (int i = tid; i < Kk * 64; i += 256) atomicAdd(&k0_acc[i], lds_k0[i]);
  if (tid == 0) atomicAdd(npos, lds_cnt);
}

// ---------------------------------------------------------------------------
// Kernel 2: row-normalize k0 -> k0n (rows 19..31 zero-padded).
// ---------------------------------------------------------------------------
__global__ void normalize_kernel(const float* __restrict__ k0,
                                 float* __restrict__ k0n) {
  const int row = threadIdx.x;  // 32 threads
  float inv = 0.0f;
  if (row < Kk) {
    float s = 0.0f;
    for (int c = 0; c < 64; ++c) { const float v = k0[row * 64 + c]; s += v * v; }
    inv = 1.0f / fmaxf(sqrtf(s), EPSV);
  }
  for (int c = 0; c < 64; ++c)
    k0n[row * 64 + c] = (row < Kk) ? k0[row * 64 + c] * inv : 0.0f;
}

// ---------------------------------------------------------------------------
// Kernel 3: per-pixel logits = (k0n . feat) / (||feat|| * tau), masked NLL.
//   A = k0n [32x64] fully in registers (2 M-tiles x 16 K-steps x float2)
//   B = feat [4ch x 16px] per K-step; sum-of-squares piggybacks on B loads.
// D layout puts pixel n in lanes n (k rows 0-7 & 16-18) and n+16 (k 8-15),
// so logsumexp/NLL complete with a single shfl_xor(.,16) pair exchange.
// USE_MASK: read the 4MB packed mask instead of re-reading 80MB of gt.
// ---------------------------------------------------------------------------
template <bool USE_MASK>
__global__ void __launch_bounds__(256)
loss_kernel(const float* __restrict__ feat, const int* __restrict__ gt,
            const unsigned int* __restrict__ wsmask,
            const float* __restrict__ k0n, float* __restrict__ loss_acc) {
  const int tid  = threadIdx.x;
  const int lane = tid & 31;
  const int half = lane >> 4;
  const int l16  = lane & 15;
  const int wave = blockIdx.x * 8 + (tid >> 5);
  const int nw   = gridDim.x * 8;

  // A tiles: lane l16 = matrix row; v0 = K 4kk+2*half, v1 = next K.
  v2f A0[16], A1[16];
#pragma unroll
  for (int kk = 0; kk < 16; ++kk) {
    const int co = 4 * kk + half * 2;
    A0[kk] = *(const v2f*)(k0n + (size_t)l16 * 64 + co);
    A1[kk] = *(const v2f*)(k0n + (size_t)(16 + l16) * 64 + co);
  }

  float acc = 0.0f;
  const int ng = NPIX / 16;          // 16-pixel groups
  for (int g = wave; g < ng; g += nw) {
    const int pg  = g * 16;
    const int b   = pg / HW;
    const int pin = (pg - b * HW) + l16;      // this lane's pixel in batch
    const float* fb = feat + (size_t)b * Cc * HW + pin;

    v8f c0 = (v8f)0.0f, c1 = (v8f)0.0f;
    float ss = 0.0f;
#pragma unroll
    for (int kk = 0; kk < 16; ++kk) {
      const int ch = 4 * kk + half * 2;
      const float b0 = fb[(size_t)ch * HW];
      const float b1 = fb[(size_t)(ch + 1) * HW];
      ss += b0 * b0 + b1 * b1;
      v2f bv; bv.x = b0; bv.y = b1;
      c0 = wmma_f32_k4(A0[kk], bv, c0);
      c1 = wmma_f32_k4(A1[kk], bv, c1);
    }
    // full per-pixel ||feat||^2: halves hold complementary channel sets
    ss += __shfl_xor(ss, 16, 32);
    const float scale = 1.0f / (fmaxf(sqrtf(ss), EPSV) * TAUV);

    // logits owned by this lane: half0 -> k {0..7,16,17,18}; half1 -> k {8..15}
    float lg[11];
#pragma unroll
    for (int j = 0; j < 8; ++j) lg[j] = c0[j] * scale;
    lg[8]  = half ? -INFINITY : c1[0] * scale;
    lg[9]  = half ? -INFINITY : c1[1] * scale;
    lg[10] = half ? -INFINITY : c1[2] * scale;

    float m = -INFINITY;
#pragma unroll
    for (int j = 0; j < 11; ++j) m = fmaxf(m, lg[j]);
    m = fmaxf(m, __shfl_xor(m, 16, 32));
    float se = 0.0f;
#pragma unroll
    for (int j = 0; j < 11; ++j) se += expf(lg[j] - m);
    se += __shfl_xor(se, 16, 32);
    const float lse = m + logf(se);

    float contrib = 0.0f;
    if (USE_MASK) {
      const unsigned mw = wsmask[pg + l16];   // both halves read same word
#pragma unroll
      for (int j = 0; j < 11; ++j) {
        if (j < 8 || !half) {
          const int k = (!half && j < 8) ? j : 8 + j;   // half0: j>=8 -> 16..18
          if ((mw >> k) & 1u) contrib += lse - lg[j];
        }
      }
    } else {
      const int* gb = gt + (size_t)b * Kk * HW + pin;
#pragma unroll
      for (int j = 0; j < 11; ++j) {
        if (j < 8 || !half) {
          const int k = (!half && j < 8) ? j : 8 + j;
          if (gb[(size_t)k * HW] == 1) contrib += lse - lg[j];
        }
      }
    }
    acc += contrib;
  }

#pragma unroll
  for (int off = 16; off > 0; off >>= 1) acc += __shfl_xor(acc, off, 32);
  __shared__ float red[8];
  if (lane == 0) red[tid >> 5] = acc;
  __syncthreads();
  if (tid == 0) {
    float s = 0.0f;
#pragma unroll
    for (int i = 0; i < 8; ++i) s += red[i];
    atomicAdd(loss_acc, s);
  }
}

// ---------------------------------------------------------------------------
// Kernel 4: final scalar.
// ---------------------------------------------------------------------------
__global__ void finalize_kernel(const float* __restrict__ loss,
                                const unsigned int* __restrict__ npos,
                                float* __restrict__ out) {
  out[0] = loss[0] / (float)npos[0];
}

extern "C" void kernel_launch(void* const* d_in, const int* in_sizes, int n_in,
                              void* d_out, int out_size, void* d_ws, size_t ws_size,
                              hipStream_t stream) {
  const float* feat = (const float*)d_in[0];
  const int*   gt   = (const int*)d_in[1];
  float* ws = (float*)d_ws;

  // ws layout (floats): [0..2047] k0 accum (19x64 used, padded),
  // [2048..4095] k0n (32x64, rows>=19 zero), [4096] num_pos (u32),
  // [4097] loss, [4352..4352+NPIX) packed gt mask (u32/pixel) if it fits.
  float*        k0    = ws;
  float*        k0n   = ws + 2048;
  unsigned int* npos  = (unsigned int*)(ws + 4096);
  float*        loss  = ws + 4097;
  unsigned int* wmask = (unsigned int*)(ws + 4352);

  const bool use_mask =
      ws_size >= (size_t)(4352 + NPIX) * sizeof(float);

  zero_kernel<<<(4352 + 255) / 256, 256, 0, stream>>>(ws, 4352);
  if (use_mask) {
    proto_kernel<true><<<512, 256, 0, stream>>>(feat, gt, k0, npos, wmask);
    normalize_kernel<<<1, 32, 0, stream>>>(k0, k0n);
    loss_kernel<true><<<2048, 256, 0, stream>>>(feat, gt, wmask, k0n, loss);
  } else {
    proto_kernel<false><<<512, 256, 0, stream>>>(feat, gt, k0, npos, wmask);
    normalize_kernel<<<1, 32, 0, stream>>>(k0, k0n);
    loss_kernel<false><<<2048, 256, 0, stream>>>(feat, gt, wmask, k0n, loss);
  }
  finalize_kernel<<<1, 1, 0, stream>>>(loss, npos, (float*)d_out);
}